// BQWarp_62732292325639
// MI455X (gfx1250) — compile-verified
//
#include <hip/hip_runtime.h>

// Ball query (BQWarp): N=32768 queries, G=4096 db points, R=0.25, K=10.
// d2 = ||q||^2 - 2 q.d + ||d||^2 per 16x16 tile via V_WMMA_F32_16X16X4_F32;
// first-K selection via ballot + popcount prefix with fully-scalar per-query
// counts; wave-uniform early exit once all 16 queries of the wave are full.

#define NQ   32768
#define GPTS 4096
#define KNB  10
#define RAD2 0.0625f

typedef __attribute__((ext_vector_type(2))) float v2f;
typedef __attribute__((ext_vector_type(8))) float v8f;

__device__ __forceinline__ float readlane_f32(float v, int lane) {
    return __int_as_float(__builtin_amdgcn_readlane(__float_as_int(v), lane));
}

__global__ __launch_bounds__(256) void bqwarp_wmma_kernel(
    const float* __restrict__ x,        // (NQ, 3)
    const float* __restrict__ grid,     // (GPTS, 3)
    float* __restrict__ map_out,        // (NQ, KNB)   indices as float
    float* __restrict__ coord_out)      // (NQ, KNB, 3)
{
    const int lane  = threadIdx.x & 31;
    const int wave  = (int)((blockIdx.x * blockDim.x + threadIdx.x) >> 5);
    const int qbase = wave * 16;
    if (qbase >= NQ) return;

    // ---- zero-prefill this wave's output region (invalid slots -> 0) ----
    for (int i = lane; i < 16 * KNB; i += 32)
        map_out[qbase * KNB + i] = 0.0f;
    for (int i = lane; i < 16 * KNB * 3; i += 32)
        coord_out[qbase * KNB * 3 + i] = 0.0f;

    // ---- load 16 query points, duplicated across half-waves ----
    const int   m16 = lane & 15;
    const bool  lo  = lane < 16;
    const float qx  = x[(qbase + m16) * 3 + 0];
    const float qy  = x[(qbase + m16) * 3 + 1];
    const float qz  = x[(qbase + m16) * 3 + 2];
    const float qq  = qx * qx + qy * qy + qz * qz;

    // A (16x4 f32) row m = [qx qy qz 1]; VGPR0 = K0|K2, VGPR1 = K1|K3
    v2f a;
    a[0] = lo ? qx : qz;
    a[1] = lo ? qy : 1.0f;

    // C = ||q||^2 arranged per 16x16 f32 C-layout (VGPR r -> M=r / M=r+8).
    // readlane(constant) gives uniform qq of query r / r+8; select per half.
    v8f cacc;
#pragma unroll
    for (int r = 0; r < 8; ++r) {
        const float qqA = readlane_f32(qq, r);      // query r   (uniform)
        const float qqB = readlane_f32(qq, r + 8);  // query r+8 (uniform)
        cacc[r] = lo ? qqA : qqB;
    }

    // Per-query hit counts: wave-uniform scalars (fed only by ballot masks).
    int cnts[16];
#pragma unroll
    for (int m = 0; m < 16; ++m) cnts[m] = 0;

    const unsigned prefix_mask = (1u << m16) - 1u;

    const int ntiles = GPTS / 16;
    for (int t = 0; t < ntiles; ++t) {
        const int gbase = t * 16;
        if (t + 1 < ntiles)
            __builtin_prefetch(grid + (gbase + 16) * 3, 0, 1);

        // per-lane db point n = lane&15 of this tile
        const float dx = grid[(gbase + m16) * 3 + 0];
        const float dy = grid[(gbase + m16) * 3 + 1];
        const float dz = grid[(gbase + m16) * 3 + 2];
        const float dd = dx * dx + dy * dy + dz * dz;

        // B (4x16 f32) col n = [-2dx -2dy -2dz dd]
        v2f b;
        b[0] = lo ? (-2.0f * dx) : (-2.0f * dz);
        b[1] = lo ? (-2.0f * dy) : dd;

        // d2[m][n] = ||q_m||^2 - 2 q_m.d_n + ||d_n||^2
        v8f d2 = __builtin_amdgcn_wmma_f32_16x16x4_f32(
            false, a, false, b, (short)0, cacc, false, false);

#pragma unroll
        for (int r = 0; r < 8; ++r) {
            const bool     hit  = d2[r] <= RAD2;
            const unsigned mask = (unsigned)__ballot(hit);   // SGPR-uniform
            const unsigned mlo  = mask & 0xFFFFu;
            const unsigned mhi  = mask >> 16;

            const int      base   = lo ? cnts[r] : cnts[r + 8]; // sgpr cndmask
            const unsigned myhalf = lo ? mlo : mhi;
            const int slot = base + __popc(myhalf & prefix_mask);

            if (hit && slot < KNB) {
                const int q = qbase + r + (lo ? 0 : 8);
                map_out[q * KNB + slot] = (float)(gbase + m16);
                float* o = coord_out + (q * KNB + slot) * 3;
                o[0] = dx; o[1] = dy; o[2] = dz;
            }

            cnts[r]     += __popc(mlo);   // s_bcnt + s_add (scalar)
            cnts[r + 8] += __popc(mhi);
        }

        // all 16 queries of this wave have K neighbors -> stop scanning
        int mn = cnts[0];
#pragma unroll
        for (int m = 1; m < 16; ++m) mn = min(mn, cnts[m]);
        if (mn >= KNB) break;             // scalar condition -> s_cbranch
    }
}

extern "C" void kernel_launch(void* const* d_in, const int* in_sizes, int n_in,
                              void* d_out, int out_size, void* d_ws, size_t ws_size,
                              hipStream_t stream) {
    (void)in_sizes; (void)n_in; (void)d_ws; (void)ws_size; (void)out_size;
    const float* x    = (const float*)d_in[0];   // (1, 32768, 3)
    const float* pg   = (const float*)d_in[1];   // (1, 32, 16, 8, 3) -> (4096, 3)
    float* map_out    = (float*)d_out;                 // first NQ*KNB elements
    float* coord_out  = (float*)d_out + NQ * KNB;      // then NQ*KNB*3 elements

    const int waves = NQ / 16;          // 2048 waves, 16 queries each
    const int block = 256;              // 8 waves per workgroup
    const int gridd = waves * 32 / block;
    bqwarp_wmma_kernel<<<gridd, block, 0, stream>>>(x, pg, map_out, coord_out);
}